// Model_1_73486890435291
// MI455X (gfx1250) — compile-verified
//
#include <hip/hip_runtime.h>
#include <math.h>

// Model dims
#define B_ 4
#define S_ 256
#define E_ 128
#define V_ 8192
#define L_ 512

typedef __attribute__((ext_vector_type(2))) float v2f;
typedef __attribute__((ext_vector_type(4))) float v4f;
typedef __attribute__((ext_vector_type(8))) float v8f;

// gcc-style int4 to exactly match the async-LDS builtin's parameter type
typedef int vi4 __attribute__((vector_size(16)));

#if defined(__AMDGCN__) && __has_builtin(__builtin_amdgcn_global_load_async_to_lds_b128)
#define HAVE_ASYNC_LDS 1
#else
#define HAVE_ASYNC_LDS 0
#endif

// ---------------------------------------------------------------------------
// Positional embedding: pe[s,c] = relu( sum_k P[s,k] * Wp[k,c] + bp[c] )
// P[s,2i] = sin(s * n^(-2i/d)), P[s,2i+1] = cos(same)
// ---------------------------------------------------------------------------
__global__ void posemb_kernel(const float* __restrict__ Wp,
                              const float* __restrict__ bp,
                              float* __restrict__ pe, int din, int dout) {
  int o = blockIdx.x * blockDim.x + threadIdx.x;
  if (o >= S_ * dout) return;
  int srow = o / dout;
  int c = o - srow * dout;
  float acc = bp[c];
  float sf = (float)srow;
  const float LN_N = 9.2103403719761836f; // ln(10000)
  for (int k = 0; k < din; ++k) {
    int i = k >> 1;
    float ang = sf * __expf(-(2.0f * (float)i / (float)din) * LN_N);
    float p = (k & 1) ? __cosf(ang) : __sinf(ang);
    acc = fmaf(p, Wp[(long)k * dout + c], acc);
  }
  pe[o] = fmaxf(acc, 0.0f);
}

// ---------------------------------------------------------------------------
// Generic WMMA f32 GEMM:  C[z] = act(A[z] @ W[z] + bias)
// 256 threads = 8 waves; each wave -> one 16x16 tile; block = one 16-row
// M tile x 128 columns. A tile double-buffered in LDS; staged with
// GLOBAL_LOAD_ASYNC_TO_LDS_B128 (ASYNCcnt) when available.
// Requirements: K % 32 == 0, N % 128 == 0. M may be ragged (ghost rows are
// pre-zeroed in LDS; EXEC is all-ones at every WMMA).
// ---------------------------------------------------------------------------
#define KC 32
#define LDSTR (KC + 4) // 36 floats = 144B row stride (16B aligned)
__global__ __launch_bounds__(256)
void wmma_gemm_kernel(const float* __restrict__ A, int lda, long strideA,
                      const float* __restrict__ W, int ldw, long strideW,
                      const float* __restrict__ bias,
                      float* __restrict__ C, int ldc, long strideC,
                      int M, int N, int K, int relu) {
  __shared__ float As[2][16][LDSTR];
  const int tid  = threadIdx.x;
  const int wave = tid >> 5; // 0..7 (wave32)
  const int lane = tid & 31;

  const int ntg = N >> 7;               // 128-column groups
  const int mt  = blockIdx.x / ntg;     // M tile
  const int ng  = blockIdx.x - mt * ntg;
  const int nt  = ng * 8 + wave;        // this wave's N tile

  const long z = blockIdx.z;
  A += z * strideA;
  W += z * strideW;
  C += z * strideC;

  const int ml    = lane & 15;          // M (A) / N (B) sub-index
  const int khalf = (lane >> 4) << 1;   // 0 for lanes 0-15, 2 for lanes 16-31
  const int ncol  = nt * 16 + ml;

  // ragged M: pre-zero both LDS buffers once; ghost rows stay zero forever
  if (M & 15) {
    for (int i = tid; i < 2 * 16 * LDSTR; i += 256) (&As[0][0][0])[i] = 0.0f;
    __syncthreads();
  }

  // stage one [16 x KC] A tile into LDS buffer `b`
  auto stage = [&](int k0, int b) {
    if (tid < 128) {               // waves 0-3 only
      int row = tid >> 3;          // 0..15
      int c4  = (tid & 7) << 2;    // 0,4,...,28
      int gr  = mt * 16 + row;
      if (gr < M) {
        const float* p = A + (long)gr * lda + k0 + c4;
#if HAVE_ASYNC_LDS
        __builtin_amdgcn_global_load_async_to_lds_b128(
            (__attribute__((address_space(1))) vi4*)(unsigned long long)(uintptr_t)p,
            (__attribute__((address_space(3))) vi4*)(unsigned)(uintptr_t)&As[b][row][c4],
            0, 0);
#else
        *(v4f*)&As[b][row][c4] = *(const v4f*)p;
        if (k0 + KC < K) __builtin_prefetch(p + KC, 0, 1);
#endif
      }
    }
  };

  v8f acc = {};
  stage(0, 0);
  int buf = 0;
  for (int k0 = 0; k0 < K; k0 += KC, buf ^= 1) {
#if HAVE_ASYNC_LDS
#if __has_builtin(__builtin_amdgcn_s_wait_asynccnt)
    __builtin_amdgcn_s_wait_asynccnt(0);
#else
    asm volatile("s_wait_asynccnt 0" ::: "memory");
#endif
#endif
    __syncthreads(); // chunk k0 resident; prior reads of buf^1 retired
    if (k0 + KC < K) {
      stage(k0 + KC, buf ^ 1); // overlap next A tile with this chunk's WMMAs
      __builtin_prefetch(W + (long)(k0 + KC) * ldw + ncol, 0, 1);
    }
#pragma unroll
    for (int kk = 0; kk < KC; kk += 4) {
      v2f a = *(const v2f*)&As[buf][ml][kk + khalf]; // A[m, k..k+1] (ISA layout)
      const float* wp = W + (long)(k0 + kk + khalf) * ldw + ncol;
      v2f b;
      b.x = wp[0];   // B[k,   n]
      b.y = wp[ldw]; // B[k+1, n]
      acc = __builtin_amdgcn_wmma_f32_16x16x4_f32(
          /*neg_a=*/false, a, /*neg_b=*/false, b,
          /*c_mod=*/(short)0, acc, /*reuse_a=*/false, /*reuse_b=*/false);
    }
  }

  // epilogue: bias + optional relu; guarded ragged-M store
  const int rowbase = mt * 16 + ((lane >> 4) << 3);
  float bv = bias ? bias[ncol] : 0.0f;
#pragma unroll
  for (int r = 0; r < 8; ++r) {
    int row = rowbase + r;
    if (row < M) {
      float v = acc[r] + bv;
      if (relu) v = fmaxf(v, 0.0f);
      C[(long)row * ldc + ncol] = v;
    }
  }
}

// ---------------------------------------------------------------------------
// Row sums: s[row] = sum_j X[row, j], one block (128 threads) per row
// ---------------------------------------------------------------------------
__global__ void rowsum_kernel(const float* __restrict__ X,
                              float* __restrict__ s, int D) {
  __shared__ float red[4];
  long row = blockIdx.x;
  const float* p = X + row * D;
  float acc = 0.0f;
  for (int j = threadIdx.x; j < D; j += blockDim.x) acc += p[j];
  for (int off = 16; off > 0; off >>= 1) acc += __shfl_down(acc, off, 32);
  if ((threadIdx.x & 31) == 0) red[threadIdx.x >> 5] = acc;
  __syncthreads();
  if (threadIdx.x == 0) s[row] = red[0] + red[1] + red[2] + red[3];
}

// ---------------------------------------------------------------------------
// M-build (the sq_mem collapse): out[b, t] = sum_k s[b,k] * W[k*T + t]
// Streams W exactly once (b128 non-temporal), all 4 batches at once.
// T % 1024 == 0 at both call sites; launched with T/1024 blocks.
// ---------------------------------------------------------------------------
__global__ __launch_bounds__(256)
void mbuild_kernel(const float* __restrict__ s, const float* __restrict__ W,
                   float* __restrict__ out, int nK, long T) {
  __shared__ float sl[B_][S_];
  for (int i = threadIdx.x; i < B_ * nK; i += blockDim.x)
    sl[i / nK][i % nK] = s[i];
  __syncthreads();
  long t4 = ((long)blockIdx.x * blockDim.x + threadIdx.x) * 4;
  v4f a0 = {}, a1 = {}, a2 = {}, a3 = {};
  const float* p = W + t4;
  for (int k = 0; k < nK; ++k) {
    v4f w = __builtin_nontemporal_load((const v4f*)(p + (long)k * T));
    a0 += sl[0][k] * w;
    a1 += sl[1][k] * w;
    a2 += sl[2][k] * w;
    a3 += sl[3][k] * w;
  }
  *(v4f*)(out + t4)         = a0;
  *(v4f*)(out + T + t4)     = a1;
  *(v4f*)(out + 2 * T + t4) = a2;
  *(v4f*)(out + 3 * T + t4) = a3;
}

// ---------------------------------------------------------------------------
// h[b,s,:] += pe[s,:]   (float4; every D is a multiple of 4)
// ---------------------------------------------------------------------------
__global__ void addpe_kernel(float* __restrict__ h, const float* __restrict__ pe,
                             int D) {
  long i4 = ((long)blockIdx.x * blockDim.x + threadIdx.x) * 4;
  if (i4 >= (long)B_ * S_ * D) return;
  long sd = i4 % ((long)S_ * D);
  v4f v = *(v4f*)(h + i4);
  v += *(const v4f*)(pe + sd);
  *(v4f*)(h + i4) = v;
}

__global__ void zero_kernel(float* __restrict__ p, int n) {
  int i = blockIdx.x * blockDim.x + threadIdx.x;
  if (i < n) p[i] = 0.0f;
}

// ---------------------------------------------------------------------------
// Split-K GEMV for flat[B,K] @ W_fd[K,512]: each block owns a 512-K chunk,
// streams W_fd once (b64 non-temporal), 4 batches at once, atomicAdd out.
// ---------------------------------------------------------------------------
#define FD_KC 512
__global__ __launch_bounds__(256)
void fd_splitk_kernel(const float* __restrict__ flat,
                      const float* __restrict__ Wfd,
                      float* __restrict__ acc, int K) {
  __shared__ float xl[B_][FD_KC];
  int k0 = blockIdx.x * FD_KC;
  for (int i = threadIdx.x; i < B_ * FD_KC; i += blockDim.x) {
    int b = i / FD_KC, kk = i - b * FD_KC;
    xl[b][kk] = flat[(long)b * K + k0 + kk];
  }
  __syncthreads();
  int n0 = threadIdx.x * 2; // two adjacent columns per thread
  v2f a[B_] = {};
  for (int kk = 0; kk < FD_KC; ++kk) {
    const float* wr = Wfd + (long)(k0 + kk) * 512;
    v2f w = __builtin_nontemporal_load((const v2f*)(wr + n0));
#pragma unroll
    for (int b = 0; b < B_; ++b) a[b] += xl[b][kk] * w;
  }
#pragma unroll
  for (int b = 0; b < B_; ++b) {
    atomicAdd(&acc[b * 512 + n0],     a[b].x);
    atomicAdd(&acc[b * 512 + n0 + 1], a[b].y);
  }
}

__global__ void bias_relu_kernel(const float* __restrict__ acc,
                                 const float* __restrict__ bias,
                                 float* __restrict__ out, int n, int D) {
  int i = blockIdx.x * blockDim.x + threadIdx.x;
  if (i < n) out[i] = fmaxf(acc[i] + bias[i % D], 0.0f);
}

// ---------------------------------------------------------------------------
static void launch_gemm(const float* A, int lda, long sA,
                        const float* W, int ldw, long sW,
                        const float* bias, float* C, int ldc, long sC,
                        int M, int N, int K, int relu, int batch,
                        hipStream_t stream) {
  int mt = (M + 15) / 16;
  dim3 grid(mt * (N >> 7), 1, batch);
  wmma_gemm_kernel<<<grid, dim3(256), 0, stream>>>(A, lda, sA, W, ldw, sW,
                                                   bias, C, ldc, sC,
                                                   M, N, K, relu);
}

extern "C" void kernel_launch(void* const* d_in, const int* in_sizes, int n_in,
                              void* d_out, int out_size, void* d_ws, size_t ws_size,
                              hipStream_t stream) {
  (void)in_sizes; (void)n_in; (void)out_size; (void)ws_size;
  const float* x     = (const float*)d_in[0];
  const float* W_emb = (const float*)d_in[1];
  const float* b_emb = (const float*)d_in[2];
  const float* Wp1   = (const float*)d_in[3];
  const float* bp1   = (const float*)d_in[4];
  const float* W_r   = (const float*)d_in[5];
  const float* b_r   = (const float*)d_in[6];
  const float* Wp2   = (const float*)d_in[7];
  const float* bp2   = (const float*)d_in[8];
  const float* W_r2  = (const float*)d_in[9];
  const float* b_r2  = (const float*)d_in[10];
  const float* Wp3   = (const float*)d_in[11];
  const float* bp3   = (const float*)d_in[12];
  const float* W_d2  = (const float*)d_in[13];
  const float* b_d2  = (const float*)d_in[14];
  const float* W_fd  = (const float*)d_in[15];
  const float* b_fd  = (const float*)d_in[16];
  const float* W1    = (const float*)d_in[17];
  const float* b1    = (const float*)d_in[18];
  const float* W2    = (const float*)d_in[19];
  const float* b2    = (const float*)d_in[20];
  const float* W3    = (const float*)d_in[21];
  const float* b3    = (const float*)d_in[22];
  const float* W_pr  = (const float*)d_in[23];
  const float* b_pr  = (const float*)d_in[24];
  float* out = (float*)d_out;

  // workspace layout (floats, 16-float aligned)
  float* ws  = (float*)d_ws;
  float* h1  = ws + 0;        // [4,256,128]   131072
  float* pe1 = ws + 131072;   // [256,128]     32768
  float* s1  = ws + 163840;   // [4,256]       1024
  float* M1  = ws + 164864;   // [4,128,256]   131072
  float* h2  = ws + 295936;   // [4,256,256]   262144
  float* pe2 = ws + 558080;   // [256,256]     65536
  float* s2  = ws + 623616;   // [4,256]       1024
  float* M2  = ws + 624640;   // [4,256,640]   655360
  float* h3  = ws + 1280000;  // [4,256,640]   655360
  float* pe3 = ws + 1935360;  // [256,640]     163840
  float* h4  = ws + 2099200;  // [4,256,512]   524288
  float* h5a = ws + 2623488;  // [4,512] acc   2048
  float* h5  = ws + 2625536;  // [4,512]
  float* h6  = ws + 2627584;
  float* h7  = ws + 2629632;
  float* h8  = ws + 2631680;

  const int twoE = 2 * E_, fiveE = 5 * E_;
  dim3 t256(256), t128(128);

  // positional embeddings (relu'd, shared across batch)
  posemb_kernel<<<(S_ * E_ + 255) / 256, t256, 0, stream>>>(Wp1, bp1, pe1, 2 * E_, E_);
  posemb_kernel<<<(S_ * twoE + 255) / 256, t256, 0, stream>>>(Wp2, bp2, pe2, 4 * E_, twoE);
  posemb_kernel<<<(S_ * fiveE + 255) / 256, t256, 0, stream>>>(Wp3, bp3, pe3, 10 * E_, fiveE);

  // h1 = relu(x @ W_emb + b_emb)   [1024,8192]@[8192,128]
  launch_gemm(x, V_, 0, W_emb, E_, 0, b_emb, h1, E_, 0,
              B_ * S_, E_, V_, 1, 1, stream);
  addpe_kernel<<<(B_ * S_ * E_ / 4 + 255) / 256, t256, 0, stream>>>(h1, pe1, E_);

  // sq_mem collapse #1
  rowsum_kernel<<<B_ * S_, t128, 0, stream>>>(h1, s1, E_);
  mbuild_kernel<<<(int)((long)E_ * twoE / 1024), t256, 0, stream>>>(s1, W_r, M1, S_, (long)E_ * twoE);
  launch_gemm(h1, E_, (long)S_ * E_, M1, twoE, (long)E_ * twoE, b_r,
              h2, twoE, (long)S_ * twoE, S_, twoE, E_, 1, B_, stream);
  addpe_kernel<<<(B_ * S_ * twoE / 4 + 255) / 256, t256, 0, stream>>>(h2, pe2, twoE);

  // sq_mem collapse #2
  rowsum_kernel<<<B_ * S_, t128, 0, stream>>>(h2, s2, twoE);
  mbuild_kernel<<<(int)((long)twoE * fiveE / 1024), t256, 0, stream>>>(s2, W_r2, M2, S_, (long)twoE * fiveE);
  launch_gemm(h2, twoE, (long)S_ * twoE, M2, fiveE, (long)twoE * fiveE, b_r2,
              h3, fiveE, (long)S_ * fiveE, S_, fiveE, twoE, 1, B_, stream);
  addpe_kernel<<<(B_ * S_ * fiveE / 4 + 255) / 256, t256, 0, stream>>>(h3, pe3, fiveE);

  // h4 = relu(h3 @ W_d2 + b_d2)   [1024,640]@[640,512]
  launch_gemm(h3, fiveE, 0, W_d2, L_, 0, b_d2, h4, L_, 0,
              B_ * S_, L_, fiveE, 1, 1, stream);

  // h5 = relu(flat @ W_fd + b_fd)  [4,131072]@[131072,512]  (split-K, atomics)
  zero_kernel<<<(B_ * L_ + 255) / 256, t256, 0, stream>>>(h5a, B_ * L_);
  fd_splitk_kernel<<<(S_ * L_) / FD_KC, t256, 0, stream>>>(h4, W_fd, h5a, S_ * L_);
  bias_relu_kernel<<<(B_ * L_ + 255) / 256, t256, 0, stream>>>(h5a, b_fd, h5, B_ * L_, L_);

  // tail MLP (M=4, ragged M handled by zero-padded LDS tiles)
  launch_gemm(h5, L_, 0, W1, L_, 0, b1, h6, L_, 0, B_, L_, L_, 1, 1, stream);
  launch_gemm(h6, L_, 0, W2, L_, 0, b2, h7, L_, 0, B_, L_, L_, 1, 1, stream);
  launch_gemm(h7, L_, 0, W3, L_, 0, b3, h8, L_, 0, B_, L_, L_, 1, 1, stream);

  // out = h8 @ W_pr + b_pr (no relu)  [4,512]@[512,8192]
  launch_gemm(h8, L_, 0, W_pr, V_, 0, b_pr, out, V_, 0, B_, V_, L_, 0, 1, stream);
}